// InvariantPointAttention_34187939676342
// MI455X (gfx1250) — compile-verified
//
#include <hip/hip_runtime.h>
#include <hip/hip_bf16.h>

// ---------------- CDNA5 fp32 WMMA (V_WMMA_F32_16X16X4_F32) -----------------
typedef __attribute__((ext_vector_type(2))) float v2f;
typedef __attribute__((ext_vector_type(8))) float v8f;

#define L_NODES 1024
#define ND      256     // NODE_DIM
#define NHEAD   4
#define NPOINT  8
#define HIDW    256     // HID
#define DHEAD   64      // HID / H
#define PTW     96      // H*P*3
#define PHD     24      // P*3
#define OCAT    352     // HID + PT

static __device__ __forceinline__ v8f wmma_f32(v2f a, v2f b, v8f c) {
  // 8 args: (neg_a, A, neg_b, B, c_mod, C, reuse_a, reuse_b)
  return __builtin_amdgcn_wmma_f32_16x16x4_f32(false, a, false, b, (short)0, c,
                                               false, false);
}

// ---------------------------------------------------------------------------
// Generic fp32 WMMA GEMM:  C[M x N] = A[M x K] * B[K x N] + bias
// block = 128 threads (4 waves), each wave owns a 16x16 tile; grid.y = M/16,
// grid.x = ceil(N/64). K must be a multiple of 4.
// A-fragment: lanes 0-15 hold (k0,k0+1) of row m=lane; lanes 16-31 (k0+2,k0+3).
// B-fragment: lane n=lane&15 holds B[k][n]; half selects k0/k0+2 row pair.
// C/D: VGPR r -> row m = r + 8*(lane>=16), col n = lane&15.
// ---------------------------------------------------------------------------
__global__ void gemm_wmma_f32_kernel(const float* __restrict__ A,
                                     const float* __restrict__ B,
                                     const float* __restrict__ bias,
                                     float* __restrict__ C,
                                     int N, int K, int lda, int ldb, int ldc) {
  const int lane = threadIdx.x & 31;
  const int wave = threadIdx.x >> 5;
  const int half = lane >> 4;
  const int l16  = lane & 15;
  const int row0 = blockIdx.y * 16;
  const int col0 = blockIdx.x * 64 + wave * 16;
  if (col0 >= N) return;  // uniform per wave

  const float* arow = A + (size_t)(row0 + l16) * lda;
  v8f acc = {};
  for (int k0 = 0; k0 < K; k0 += 4) {
    v2f a = *(const v2f*)(arow + k0 + 2 * half);
    v2f b;
    b.x = B[(size_t)(k0 + 2 * half) * ldb + col0 + l16];
    b.y = B[(size_t)(k0 + 2 * half + 1) * ldb + col0 + l16];
    acc = wmma_f32(a, b, acc);
  }
  const float bv = bias ? bias[col0 + l16] : 0.0f;
#pragma unroll
  for (int r = 0; r < 8; ++r) {
    const int m = r + 8 * half;
    C[(size_t)(row0 + m) * ldc + col0 + l16] = acc[r] + bv;
  }
}

// ---------------------------------------------------------------------------
// Transform local points to the global frame (in place) and precompute
// squared norms of q_pg / k_pg per (node, head). One thread per (l, h).
// ---------------------------------------------------------------------------
__global__ void point_frames_kernel(const float* __restrict__ rot,
                                    const float* __restrict__ trans,
                                    float* __restrict__ qp,
                                    float* __restrict__ kp,
                                    float* __restrict__ vp,
                                    float* __restrict__ qn,
                                    float* __restrict__ kn) {
  const int idx = blockIdx.x * blockDim.x + threadIdx.x;
  if (idx >= L_NODES * NHEAD) return;
  const int l = idx >> 2;
  const int h = idx & 3;
  float R[9], t[3];
#pragma unroll
  for (int i = 0; i < 9; ++i) R[i] = rot[l * 9 + i];
#pragma unroll
  for (int i = 0; i < 3; ++i) t[i] = trans[l * 3 + i];

  float nq = 0.0f, nk = 0.0f;
  const size_t base = (size_t)l * PTW + h * PHD;
#pragma unroll
  for (int p = 0; p < NPOINT; ++p) {
    const size_t o = base + p * 3;
    // q
    {
      float x0 = qp[o + 0], x1 = qp[o + 1], x2 = qp[o + 2];
#pragma unroll
      for (int i = 0; i < 3; ++i) {
        float y = R[i * 3 + 0] * x0 + R[i * 3 + 1] * x1 + R[i * 3 + 2] * x2 + t[i];
        qp[o + i] = y;
        nq += y * y;
      }
    }
    // k
    {
      float x0 = kp[o + 0], x1 = kp[o + 1], x2 = kp[o + 2];
#pragma unroll
      for (int i = 0; i < 3; ++i) {
        float y = R[i * 3 + 0] * x0 + R[i * 3 + 1] * x1 + R[i * 3 + 2] * x2 + t[i];
        kp[o + i] = y;
        nk += y * y;
      }
    }
    // v
    {
      float x0 = vp[o + 0], x1 = vp[o + 1], x2 = vp[o + 2];
#pragma unroll
      for (int i = 0; i < 3; ++i) {
        float y = R[i * 3 + 0] * x0 + R[i * 3 + 1] * x1 + R[i * 3 + 2] * x2 + t[i];
        vp[o + i] = y;
      }
    }
  }
  qn[l * NHEAD + h] = nq;
  kn[l * NHEAD + h] = nk;
}

// ---------------------------------------------------------------------------
// Fused attention per (head, 16-query tile).  256 threads = 8 waves.
// Phase 1: logits(16x1024) via WMMA; point-distance term expanded into
//          norms + dot products so it also runs on the matrix pipe.
//          Q-side A fragments are register-resident across all 8 col tiles.
// Phase 2: in-wave softmax (wave w owns rows 2w, 2w+1; 16-lane shfl reduce).
// Phase 3: attn(16x1024) x [v_s | v_pg](1024x88) via WMMA over K = 1024.
// LDS row stride padded to 1028 floats -> conflict-free fragment reads.
// ---------------------------------------------------------------------------
#define LSTRIDE 1028
__global__ void ipa_attention_kernel(const float* __restrict__ q_s,
                                     const float* __restrict__ k_s,
                                     const float* __restrict__ v_s,
                                     const float* __restrict__ q_pg,
                                     const float* __restrict__ k_pg,
                                     const float* __restrict__ v_pg,
                                     const float* __restrict__ qn,
                                     const float* __restrict__ kn,
                                     float* __restrict__ out_cat,
                                     float* __restrict__ op_g) {
  __shared__ float attn[16 * LSTRIDE];

  const int h    = blockIdx.y;
  const int i0   = blockIdx.x * 16;
  const int lane = threadIdx.x & 31;
  const int wave = threadIdx.x >> 5;
  const int half = lane >> 4;
  const int l16  = lane & 15;

  const float* qsrow = q_s + (size_t)(i0 + l16) * HIDW + h * DHEAD;
  const float* qprow = q_pg + (size_t)(i0 + l16) * PTW + h * PHD;

  // ---- register-resident A fragments for the whole query tile ----
  v2f aq[DHEAD / 4];          // scalar q, pre-scaled by 1/sqrt(D) = 0.125
#pragma unroll
  for (int s = 0; s < DHEAD / 4; ++s) {
    v2f a = *(const v2f*)(qsrow + 4 * s + 2 * half);
    a.x *= 0.125f;
    a.y *= 0.125f;
    aq[s] = a;
  }
  v2f ap[PHD / 4];            // point q (global frame)
#pragma unroll
  for (int s = 0; s < PHD / 4; ++s)
    ap[s] = *(const v2f*)(qprow + 4 * s + 2 * half);

  float qbias[8];             // -0.5*||q_pg||^2 per output row
#pragma unroll
  for (int r = 0; r < 8; ++r)
    qbias[r] = -0.5f * qn[(size_t)(i0 + r + 8 * half) * NHEAD + h];

  // ---------------- Phase 1: logits ----------------
  for (int ct = 0; ct < 8; ++ct) {
    const int col0 = wave * 128 + ct * 16;
    const float* ksrow = k_s + (size_t)(col0 + l16) * HIDW + h * DHEAD;
    const float* kprow = k_pg + (size_t)(col0 + l16) * PTW + h * PHD;
    const float knb = -0.5f * kn[(size_t)(col0 + l16) * NHEAD + h];
    v8f acc = {};
#pragma unroll
    for (int s = 0; s < DHEAD / 4; ++s) {
      v2f b = *(const v2f*)(ksrow + 4 * s + 2 * half);
      acc = wmma_f32(aq[s], b, acc);
    }
#pragma unroll
    for (int s = 0; s < PHD / 4; ++s) {
      v2f b = *(const v2f*)(kprow + 4 * s + 2 * half);
      acc = wmma_f32(ap[s], b, acc);
    }
#pragma unroll
    for (int r = 0; r < 8; ++r) {
      const int m = r + 8 * half;
      attn[m * LSTRIDE + col0 + l16] = acc[r] + qbias[r] + knb;
    }
  }
  __syncthreads();

  // ---------------- Phase 2: softmax (normalized in place) ----------------
  {
    const int row = 2 * wave + half;           // lanes 0-15: row 2w, 16-31: 2w+1
    float* rowp = attn + row * LSTRIDE;
    float rmax = -1e30f;
    for (int t = 0; t < 64; ++t) rmax = fmaxf(rmax, rowp[l16 + 16 * t]);
    for (int off = 8; off; off >>= 1) rmax = fmaxf(rmax, __shfl_xor(rmax, off, 32));
    float s = 0.0f;
    for (int t = 0; t < 64; ++t) {
      const float e = __expf(rowp[l16 + 16 * t] - rmax);
      rowp[l16 + 16 * t] = e;
      s += e;
    }
    for (int off = 8; off; off >>= 1) s += __shfl_xor(s, off, 32);
    const float inv = 1.0f / s;
    for (int t = 0; t < 64; ++t) rowp[l16 + 16 * t] *= inv;
  }
  __syncthreads();

  // ---------------- Phase 3: attn @ [v_s | v_pg] ----------------
  if (wave < 6) {                               // 6 tiles cover 88 (pad to 96) cols
    const int colg = wave * 16 + l16;           // 0..95
    const float* vbase = v_s;
    int vstride = 0;
    bool valid = true;
    if (colg < DHEAD) {
      vbase = v_s + h * DHEAD + colg;  vstride = HIDW;
    } else if (colg < DHEAD + PHD) {
      vbase = v_pg + h * PHD + (colg - DHEAD);  vstride = PTW;
    } else {
      valid = false;
    }
    v8f acc = {};
    for (int k0 = 0; k0 < L_NODES; k0 += 4) {
      v2f a = *(const v2f*)(attn + l16 * LSTRIDE + k0 + 2 * half);
      const int kk = k0 + 2 * half;
      v2f b;
      b.x = valid ? vbase[(size_t)kk * vstride] : 0.0f;
      b.y = valid ? vbase[(size_t)(kk + 1) * vstride] : 0.0f;
      acc = wmma_f32(a, b, acc);
    }
#pragma unroll
    for (int r = 0; r < 8; ++r) {
      const int i = i0 + r + 8 * half;
      if (colg < DHEAD)
        out_cat[(size_t)i * OCAT + h * DHEAD + colg] = acc[r];
      else if (colg < DHEAD + PHD)
        op_g[(size_t)i * PTW + h * PHD + (colg - DHEAD)] = acc[r];
    }
  }
}

// ---------------------------------------------------------------------------
// out_pl = R^T (out_p - t); pack into out_cat columns [256, 352).
// ---------------------------------------------------------------------------
__global__ void finalize_points_kernel(const float* __restrict__ rot,
                                       const float* __restrict__ trans,
                                       const float* __restrict__ op_g,
                                       float* __restrict__ out_cat) {
  const int idx = blockIdx.x * blockDim.x + threadIdx.x;
  if (idx >= L_NODES * NHEAD) return;
  const int l = idx >> 2;
  const int h = idx & 3;
  float R[9], t[3];
#pragma unroll
  for (int i = 0; i < 9; ++i) R[i] = rot[l * 9 + i];
#pragma unroll
  for (int i = 0; i < 3; ++i) t[i] = trans[l * 3 + i];

#pragma unroll
  for (int p = 0; p < NPOINT; ++p) {
    const size_t o = (size_t)l * PTW + h * PHD + p * 3;
    const float x0 = op_g[o + 0] - t[0];
    const float x1 = op_g[o + 1] - t[1];
    const float x2 = op_g[o + 2] - t[2];
#pragma unroll
    for (int i = 0; i < 3; ++i) {
      const float y = R[0 * 3 + i] * x0 + R[1 * 3 + i] * x1 + R[2 * 3 + i] * x2;
      out_cat[(size_t)l * OCAT + HIDW + h * PHD + p * 3 + i] = y;
    }
  }
}

// ---------------------------------------------------------------------------
extern "C" void kernel_launch(void* const* d_in, const int* in_sizes, int n_in,
                              void* d_out, int out_size, void* d_ws, size_t ws_size,
                              hipStream_t stream) {
  const float* nf    = (const float*)d_in[0];
  const float* trans = (const float*)d_in[1];
  const float* rot   = (const float*)d_in[2];
  const float* Wq  = (const float*)d_in[3];
  const float* bq  = (const float*)d_in[4];
  const float* Wk  = (const float*)d_in[5];
  const float* bk  = (const float*)d_in[6];
  const float* Wv  = (const float*)d_in[7];
  const float* bv  = (const float*)d_in[8];
  const float* Wqp = (const float*)d_in[9];
  const float* bqp = (const float*)d_in[10];
  const float* Wkp = (const float*)d_in[11];
  const float* bkp = (const float*)d_in[12];
  const float* Wvp = (const float*)d_in[13];
  const float* bvp = (const float*)d_in[14];
  const float* Wo  = (const float*)d_in[15];
  const float* bo  = (const float*)d_in[16];
  float* out = (float*)d_out;

  // workspace layout (floats); total ~6.2 MB
  float* ws  = (float*)d_ws;
  float* q_s = ws;                                 // 1024*256
  float* k_s = q_s + L_NODES * HIDW;
  float* v_s = k_s + L_NODES * HIDW;
  float* q_p = v_s + L_NODES * HIDW;               // 1024*96 (becomes q_pg)
  float* k_p = q_p + L_NODES * PTW;
  float* v_p = k_p + L_NODES * PTW;
  float* qn  = v_p + L_NODES * PTW;                // 1024*4
  float* kn  = qn + L_NODES * NHEAD;
  float* opg = kn + L_NODES * NHEAD;               // 1024*96
  float* oc  = opg + L_NODES * PTW;                // 1024*352

  const dim3 blkG(128);
  const dim3 grdS(4, L_NODES / 16);   // N = 256
  const dim3 grdP(2, L_NODES / 16);   // N = 96

  // projections
  gemm_wmma_f32_kernel<<<grdS, blkG, 0, stream>>>(nf, Wq,  bq,  q_s, HIDW, ND, ND, HIDW, HIDW);
  gemm_wmma_f32_kernel<<<grdS, blkG, 0, stream>>>(nf, Wk,  bk,  k_s, HIDW, ND, ND, HIDW, HIDW);
  gemm_wmma_f32_kernel<<<grdS, blkG, 0, stream>>>(nf, Wv,  bv,  v_s, HIDW, ND, ND, HIDW, HIDW);
  gemm_wmma_f32_kernel<<<grdP, blkG, 0, stream>>>(nf, Wqp, bqp, q_p, PTW,  ND, ND, PTW,  PTW);
  gemm_wmma_f32_kernel<<<grdP, blkG, 0, stream>>>(nf, Wkp, bkp, k_p, PTW,  ND, ND, PTW,  PTW);
  gemm_wmma_f32_kernel<<<grdP, blkG, 0, stream>>>(nf, Wvp, bvp, v_p, PTW,  ND, ND, PTW,  PTW);

  // local -> global frames + norms
  point_frames_kernel<<<dim3(16), dim3(256), 0, stream>>>(rot, trans, q_p, k_p, v_p, qn, kn);

  // fused attention
  ipa_attention_kernel<<<dim3(L_NODES / 16, NHEAD), dim3(256), 0, stream>>>(
      q_s, k_s, v_s, q_p, k_p, v_p, qn, kn, oc, opg);

  // rotate back + pack
  finalize_points_kernel<<<dim3(16), dim3(256), 0, stream>>>(rot, trans, opg, oc);

  // output projection: (1024 x 352) @ (352 x 256) + bo
  gemm_wmma_f32_kernel<<<grdS, blkG, 0, stream>>>(oc, Wo, bo, out, HIDW, OCAT, OCAT, HIDW, ND);
}